// CondNMSPostProcess_20486994002101
// MI455X (gfx1250) — compile-verified
//
#include <hip/hip_runtime.h>
#include <stdint.h>
#include <math.h>

// ---------------------------------------------------------------------------
// CondNMSPostProcess for MI455X (gfx1250, wave32).
// One workgroup of 128 threads (4 wave32) per (batch, patch).
//   Phase A: async global->LDS staging (global_load_async_to_lds_*, ASYNCcnt)
//   Phase B: sigmoid + cxcywh->xyxy*scale, in LDS
//   Phase C: 1024-entry bitonic sort (desc) of (score, index)
//   Phase D: gather top-100 boxes/areas
//   Phase E: v_wmma_f32_16x16x4_f32 pairwise area-sum matrix (112x112)
//   Phase F: greedy NMS, 1 candidate per thread, serial over ranks
//   Phase G: top-20 selection emulating jax top_k -inf tie-break
//   Phase H: write (score, x1,y1,x2,y2) records
// ---------------------------------------------------------------------------

typedef __attribute__((ext_vector_type(2))) float v2f;
typedef __attribute__((ext_vector_type(8))) float v8f;

#define NQ        900
#define NPAD      1024
#define TOPK_PRE  100
#define NT        112      // TOPK_PRE padded to 7 * 16 for WMMA tiling
#define TOPK_POST 20
#define IOU_THR   0.7f
#define BLOCK     128

// Low 32 bits of a flat shared-memory pointer are the wave-relative LDS byte
// address (ISA 10.2: LDS aperture).
__device__ __forceinline__ unsigned lds_addr32(const void* p) {
  return (unsigned)(unsigned long long)(uintptr_t)p;
}

__global__ __launch_bounds__(BLOCK)
void cond_nms_pp_kernel(const float* __restrict__ logits,
                        const float* __restrict__ boxes,
                        const float* __restrict__ tsizes,
                        float* __restrict__ out,
                        int N) {
  // Aliased region:
  //   phases A-D: s_box[1024*4] (16384 B) | s_key[1024] (4096 B) | s_idx[1024] (4096 B)
  //   phases E-F: s_sum[112*112] (50176 B)
  __shared__ __align__(16) char smem[50176];
  float* s_box = (float*)smem;            // [NPAD*4] raw then xyxy boxes
  float* s_key = (float*)(smem + 16384);  // [NPAD]   raw logit ch3 then score
  int*   s_idx = (int*)  (smem + 20480);  // [NPAD]   original query index
  float* s_sum = (float*)smem;            // [NT*NT]  area[i] + area[j]

  __shared__ float s_nbox[NT * 4];
  __shared__ float s_nscore[NT];
  __shared__ float s_area[NT];
  __shared__ int   s_keep[NT];
  __shared__ int   s_sel[TOPK_POST];

  const int t  = threadIdx.x;
  const int wg = blockIdx.x;              // wg = b * P + p
  const int P  = N / NQ;
  const int b  = wg / P;
  const int p  = wg % P;

  const unsigned long long base = (unsigned long long)b * (unsigned long long)N
                                + (unsigned long long)p * NQ;

  // -------- Phase A: async global -> LDS staging (gfx1250 async path) ------
  for (int q = t; q < NQ; q += BLOCK) {
    // box: 16 bytes (cx,cy,w,h) -> s_box[q*4..q*4+3]
    unsigned long long gb = (unsigned long long)(const void*)(boxes + (base + q) * 4);
    unsigned db = lds_addr32(&s_box[q * 4]);
    asm volatile("global_load_async_to_lds_b128 %0, %1, off"
                 :: "v"(db), "v"(gb) : "memory");
    // logit channel 3: 4 bytes -> s_key[q]
    unsigned long long gl = (unsigned long long)(const void*)(logits + (base + q) * 4 + 3);
    unsigned dl = lds_addr32(&s_key[q]);
    asm volatile("global_load_async_to_lds_b32 %0, %1, off"
                 :: "v"(dl), "v"(gl) : "memory");
  }
  // pad sort keys [900,1024)
  for (int q = NQ + t; q < NPAD; q += BLOCK) {
    s_key[q] = -INFINITY;
    s_idx[q] = q;
  }
  asm volatile("s_wait_asynccnt 0" ::: "memory");
  __syncthreads();

  // -------- Phase B: transform in LDS --------------------------------------
  const float ts0 = tsizes[b * 2 + 0];   // h
  const float ts1 = tsizes[b * 2 + 1];   // w ; scale = (w, h, w, h)
  for (int q = t; q < NQ; q += BLOCK) {
    float x  = s_key[q];
    s_key[q] = 1.0f / (1.0f + __expf(-x));       // sigmoid
    s_idx[q] = q;
    float cx = s_box[q * 4 + 0], cy = s_box[q * 4 + 1];
    float w  = s_box[q * 4 + 2], h  = s_box[q * 4 + 3];
    s_box[q * 4 + 0] = (cx - 0.5f * w) * ts1;
    s_box[q * 4 + 1] = (cy - 0.5f * h) * ts0;
    s_box[q * 4 + 2] = (cx + 0.5f * w) * ts1;
    s_box[q * 4 + 3] = (cy + 0.5f * h) * ts0;
  }
  __syncthreads();

  // -------- Phase C: bitonic sort 1024, descending by score ----------------
  for (int k = 2; k <= NPAD; k <<= 1) {
    for (int j = k >> 1; j > 0; j >>= 1) {
      for (int i = t; i < NPAD; i += BLOCK) {
        int l = i ^ j;
        if (l > i) {
          float ki = s_key[i], kl = s_key[l];
          bool descBlock = ((i & k) == 0);
          bool doswap = descBlock ? (ki < kl) : (ki > kl);
          if (doswap) {
            s_key[i] = kl; s_key[l] = ki;
            int tmp = s_idx[i]; s_idx[i] = s_idx[l]; s_idx[l] = tmp;
          }
        }
      }
      __syncthreads();
    }
  }

  // -------- Phase D: gather top-100 (sorted) -------------------------------
  if (t < NT) {
    if (t < TOPK_PRE) {
      int qi = s_idx[t];
      float x0 = s_box[qi * 4 + 0], y0 = s_box[qi * 4 + 1];
      float x1 = s_box[qi * 4 + 2], y1 = s_box[qi * 4 + 3];
      s_nbox[t * 4 + 0] = x0; s_nbox[t * 4 + 1] = y0;
      s_nbox[t * 4 + 2] = x1; s_nbox[t * 4 + 3] = y1;
      s_nscore[t] = s_key[t];
      s_area[t]   = (x1 - x0) * (y1 - y0);
      s_keep[t]   = 1;
    } else {
      s_nbox[t * 4 + 0] = 0.f; s_nbox[t * 4 + 1] = 0.f;
      s_nbox[t * 4 + 2] = 0.f; s_nbox[t * 4 + 3] = 0.f;
      s_nscore[t] = -INFINITY;
      s_area[t]   = 0.f;
      s_keep[t]   = 0;
    }
  }
  __syncthreads();   // also fences before s_sum aliases s_box/s_key/s_idx

  // -------- Phase E: WMMA pairwise area sums: D[m][n] = area_m + area_n ----
  // A(16x4) = [area_i, 1, 0, 0] ; B(4x16) = [1; area_j; 0; 0]
  {
    const int wave = t >> 5;
    const int lane = t & 31;
    for (int tile = wave; tile < 49; tile += (BLOCK / 32)) {
      int ti = tile / 7, tj = tile % 7;
      v2f A, Bm;
      if (lane < 16) {                    // lanes 0-15: K=0 (v0), K=1 (v1)
        A.x  = s_area[ti * 16 + lane];    // A[m][0] = area_i
        A.y  = 1.0f;                      // A[m][1] = 1
        Bm.x = 1.0f;                      // B[0][n] = 1
        Bm.y = s_area[tj * 16 + lane];    // B[1][n] = area_j
      } else {                            // lanes 16-31: K=2,3 rows -> zero
        A.x = 0.f; A.y = 0.f; Bm.x = 0.f; Bm.y = 0.f;
      }
      v8f C = {0.f, 0.f, 0.f, 0.f, 0.f, 0.f, 0.f, 0.f};
      v8f D = __builtin_amdgcn_wmma_f32_16x16x4_f32(
          /*neg_a=*/false, A, /*neg_b=*/false, Bm,
          /*c_mod=*/(short)0, C, /*reuse_a=*/false, /*reuse_b=*/false);
      int m0 = (lane < 16) ? 0 : 8;
      int n  = (lane < 16) ? lane : (lane - 16);
#pragma unroll
      for (int r = 0; r < 8; ++r)
        s_sum[(ti * 16 + m0 + r) * NT + (tj * 16 + n)] = D[r];
    }
  }
  __syncthreads();

  // -------- Phase F: greedy NMS (one candidate per thread) -----------------
  for (int i = 0; i < TOPK_PRE - 1; ++i) {
    if (t > i && t < TOPK_PRE) {
      if (s_keep[i] && s_keep[t]) {
        float lt0 = fmaxf(s_nbox[i * 4 + 0], s_nbox[t * 4 + 0]);
        float lt1 = fmaxf(s_nbox[i * 4 + 1], s_nbox[t * 4 + 1]);
        float rb0 = fminf(s_nbox[i * 4 + 2], s_nbox[t * 4 + 2]);
        float rb1 = fminf(s_nbox[i * 4 + 3], s_nbox[t * 4 + 3]);
        float w = fmaxf(rb0 - lt0, 0.f);
        float h = fmaxf(rb1 - lt1, 0.f);
        float inter = w * h;
        float iou = inter / (s_sum[i * NT + t] - inter);
        if (iou > IOU_THR) s_keep[t] = 0;
      }
    }
    __syncthreads();
  }

  // -------- Phase G: top-20 selection (emulates top_k -inf tie-break) ------
  if (t == 0) {
    int cnt = 0;
    for (int i = 0; i < TOPK_PRE && cnt < TOPK_POST; ++i)
      if (s_keep[i]) s_sel[cnt++] = i;
    for (int i = 0; i < TOPK_PRE && cnt < TOPK_POST; ++i)
      if (!s_keep[i]) s_sel[cnt++] = i;
  }
  __syncthreads();

  // -------- Phase H: write output records (score, x1, y1, x2, y2) ----------
  if (t < TOPK_POST) {
    int sel = s_sel[t];
    unsigned long long o =
        ((unsigned long long)wg * TOPK_POST + (unsigned long long)t) * 5ull;
    out[o + 0] = s_nscore[sel];
    out[o + 1] = s_nbox[sel * 4 + 0];
    out[o + 2] = s_nbox[sel * 4 + 1];
    out[o + 3] = s_nbox[sel * 4 + 2];
    out[o + 4] = s_nbox[sel * 4 + 3];
  }
}

extern "C" void kernel_launch(void* const* d_in, const int* in_sizes, int n_in,
                              void* d_out, int out_size, void* d_ws, size_t ws_size,
                              hipStream_t stream) {
  const float* logits = (const float*)d_in[0];   // (B, N, 4) f32
  const float* boxes  = (const float*)d_in[1];   // (B, N, 4) f32
  const float* tsz    = (const float*)d_in[2];   // (B, 2)    f32
  // d_in[3] is num_queries on device; fixed at 900 by the harness setup.
  float* out = (float*)d_out;                    // (B, P*20, 5) f32

  const int B = in_sizes[2] / 2;
  const int N = in_sizes[1] / (4 * B);           // 57600
  const int P = N / NQ;                          // 64

  dim3 grid(B * P);
  dim3 block(BLOCK);
  cond_nms_pp_kernel<<<grid, block, 0, stream>>>(logits, boxes, tsz, out, N);
}